// GAT_60533269069991
// MI455X (gfx1250) — compile-verified
//
#include <hip/hip_runtime.h>

typedef float v2f __attribute__((ext_vector_type(2)));
typedef float v8f __attribute__((ext_vector_type(8)));

#define NEG_SLOPE 0.2f

// ---------------------------------------------------------------------------
// GEMM: Y[M, Ncols] = X[M, K] @ Wp[K, NT*16], fp32, V_WMMA_F32_16X16X4_F32.
// One wave32 owns a 16-row strip and ALL NT column tiles: A tile is loaded
// once per K-step and reused across NT WMMAs (guard-free inner loop; Wp is
// zero-padded to a multiple of 16 columns). Store epilogue guards col<Ncols.
// A layout (16x4 f32): lanes 0-15 hold M=lane, K={k,k+1}; lanes 16-31 hold
// M=lane-16, K={k+2,k+3}. B layout (4x16): VGPR0 lanes0-15 = row k,
// lanes16-31 = row k+2; VGPR1 = rows k+1 / k+3. C/D: VGPR r -> M=r (+8 for
// upper half-lanes), N=lane&15.
// ---------------------------------------------------------------------------
template <int NT>
__global__ __launch_bounds__(128) void gat_gemm_wmma(
    const float* __restrict__ X, const float* __restrict__ Wp,
    float* __restrict__ Y, int M, int K, int Ncols)
{
    const int wave = blockIdx.x * (blockDim.x >> 5) + (threadIdx.x >> 5);
    if (wave >= (M >> 4)) return;

    const int lane = threadIdx.x & 31;
    const int l16  = lane & 15;
    const int half = lane >> 4;
    const int row  = wave * 16 + l16;
    const int NP   = NT * 16;            // padded B/W stride

    v8f acc[NT];
    #pragma unroll
    for (int t = 0; t < NT; ++t) acc[t] = (v8f){};

    for (int k = 0; k < K; k += 4) {
        const int kk = k + 2 * half;
        // A: two consecutive K elements of one X row (8B aligned: K % 4 == 0)
        v2f a = *(const v2f*)(X + (size_t)row * K + kk);
        const float* w0 = Wp + (size_t)kk * NP + l16;
        const float* w1 = w0 + NP;
        #pragma unroll
        for (int t = 0; t < NT; ++t) {
            v2f b;
            b.x = w0[t * 16];
            b.y = w1[t * 16];
            acc[t] = __builtin_amdgcn_wmma_f32_16x16x4_f32(
                /*neg_a=*/false, a, /*neg_b=*/false, b,
                /*c_mod=*/(short)0, acc[t], /*reuse_a=*/false, /*reuse_b=*/false);
        }
    }

    #pragma unroll
    for (int t = 0; t < NT; ++t) {
        const int col = t * 16 + l16;
        if (col < Ncols) {
            #pragma unroll
            for (int r = 0; r < 8; ++r) {
                const int orow = wave * 16 + r + 8 * half;
                Y[(size_t)orow * Ncols + col] = acc[t][r];
            }
        }
    }
}

// Zero-pad W [K, Ncols] -> Wp [K, NP] (NP multiple of 16).
__global__ void gat_pad_w(const float* __restrict__ W, float* __restrict__ Wp,
                          int K, int Ncols, int NP)
{
    int i = blockIdx.x * blockDim.x + threadIdx.x;
    if (i >= K * NP) return;
    int k = i / NP, c = i - k * NP;
    Wp[i] = (c < Ncols) ? W[(size_t)k * Ncols + c] : 0.0f;
}

// el[n,h] = sum_d feat[n,h,d]*al[h,d];  er likewise with ar.
__global__ void gat_node_attn(const float* __restrict__ feat,
                              const float* __restrict__ al,
                              const float* __restrict__ ar,
                              float* __restrict__ el, float* __restrict__ er,
                              int Nn, int H, int D)
{
    int i = blockIdx.x * blockDim.x + threadIdx.x;   // n*H + h
    if (i >= Nn * H) return;
    int n = i / H, h = i - n * H;
    const float* f  = feat + (size_t)n * H * D + (size_t)h * D;
    const float* pl = al + (size_t)h * D;
    const float* pr = ar + (size_t)h * D;
    float sl = 0.f, sr = 0.f;
    for (int d = 0; d < D; ++d) { float v = f[d]; sl += v * pl[d]; sr += v * pr[d]; }
    el[i] = sl; er[i] = sr;
}

// m = -inf, s = 0 (first NH elems), acc = 0 (NHD elems).
__global__ void gat_init(float* __restrict__ m, float* __restrict__ s,
                         float* __restrict__ acc, int NH, int NHD)
{
    int i = blockIdx.x * blockDim.x + threadIdx.x;
    if (i < NH)  { m[i] = -__builtin_inff(); s[i] = 0.f; }
    if (i < NHD) acc[i] = 0.f;
}

// float atomic max via order-preserving integer mapping (mixed-sign safe).
__device__ __forceinline__ void atomicMaxF(float* addr, float v)
{
    if (v >= 0.f)
        atomicMax((int*)addr, __float_as_int(v));
    else
        atomicMin((unsigned int*)addr, __float_as_uint(v));
}

// Pass A: per (edge, head) raw score -> segment max over dst.
__global__ void gat_edge_max(const float* __restrict__ el,
                             const float* __restrict__ er,
                             const int* __restrict__ src,
                             const int* __restrict__ dst,
                             float* __restrict__ m, int E, int H)
{
    int i = blockIdx.x * blockDim.x + threadIdx.x;   // e*H + h
    if (i >= E * H) return;
    int e = i / H, h = i - e * H;
    int s = src[e], d = dst[e];
    float x = el[s * H + h] + er[d * H + h];
    x = (x > 0.f) ? x : NEG_SLOPE * x;
    atomicMaxF(&m[d * H + h], x);
}

// Pass B: w = exp(e - m[dst]); store w, accumulate Sum(w) per dst.
__global__ void gat_edge_w(const float* __restrict__ el,
                           const float* __restrict__ er,
                           const float* __restrict__ m,
                           const int* __restrict__ src,
                           const int* __restrict__ dst,
                           float* __restrict__ wbuf,
                           float* __restrict__ ssum, int E, int H)
{
    int i = blockIdx.x * blockDim.x + threadIdx.x;   // e*H + h
    if (i >= E * H) return;
    int e = i / H, h = i - e * H;
    int s = src[e], d = dst[e];
    float x = el[s * H + h] + er[d * H + h];
    x = (x > 0.f) ? x : NEG_SLOPE * x;
    float w = __expf(x - m[d * H + h]);
    wbuf[i] = w;
    atomicAdd(&ssum[d * H + h], w);
}

// Pass C: unnormalized aggregation acc[dst,h,d] += w[e,h]*feat[src,h,d].
// One thread per (edge, d): lanes of a wave cover consecutive d -> coalesced
// feat reads and coalesced global_atomic_add_f32.
__global__ void gat_edge_agg(const float* __restrict__ wbuf,
                             const float* __restrict__ feat,
                             const int* __restrict__ src,
                             const int* __restrict__ dst,
                             float* __restrict__ acc, int E, int H, int D)
{
    long long i = (long long)blockIdx.x * blockDim.x + threadIdx.x;
    if (i >= (long long)E * D) return;
    int e = (int)(i / D);
    int d = (int)(i - (long long)e * D);
    int s = src[e], t = dst[e];
    const float* fs = feat + (size_t)s * H * D;
    float*       at = acc  + (size_t)t * H * D;
    for (int h = 0; h < H; ++h) {
        float w = wbuf[(size_t)e * H + h];
        atomicAdd(&at[h * D + d], w * fs[h * D + d]);
    }
}

// Pass D: out = acc / Sum(w) + bias, optional ReLU.
__global__ void gat_node_finish(const float* __restrict__ acc,
                                const float* __restrict__ ssum,
                                const float* __restrict__ bias,
                                float* __restrict__ out,
                                int Nn, int H, int D, int relu)
{
    int i = blockIdx.x * blockDim.x + threadIdx.x;   // n*H*D + h*D + d
    if (i >= Nn * H * D) return;
    int d  = i % D;
    int nh = i / D;
    int h  = nh % H;
    float v = acc[i] / ssum[nh] + bias[h * D + d];
    if (relu) v = (v > 0.f) ? v : 0.f;
    out[i] = v;
}

extern "C" void kernel_launch(void* const* d_in, const int* in_sizes, int n_in,
                              void* d_out, int out_size, void* d_ws, size_t ws_size,
                              hipStream_t stream)
{
    const float* x   = (const float*)d_in[0];
    const int*   src = (const int*)  d_in[1];
    const int*   dst = (const int*)  d_in[2];
    const float* W0  = (const float*)d_in[3];
    const float* al0 = (const float*)d_in[4];
    const float* ar0 = (const float*)d_in[5];
    const float* b0  = (const float*)d_in[6];
    const float* W1  = (const float*)d_in[7];
    const float* al1 = (const float*)d_in[8];
    const float* ar1 = (const float*)d_in[9];
    const float* b1  = (const float*)d_in[10];
    const float* W2  = (const float*)d_in[11];
    const float* al2 = (const float*)d_in[12];
    const float* ar2 = (const float*)d_in[13];
    const float* b2  = (const float*)d_in[14];

    const int Nn = in_sizes[0] / 256;   // 50000
    const int E  = in_sizes[1];         // 1.6M

    // Workspace carve-up (floats).
    float* ws = (float*)d_ws;
    size_t o = 0;
    float* featA = ws + o; o += (size_t)Nn * 128;   // GEMM output / edge source
    float* featB = ws + o; o += (size_t)Nn * 128;   // layer output / next input
    float* accb  = ws + o; o += (size_t)Nn * 128;   // unnormalized aggregation
    float* wbuf  = ws + o; o += (size_t)E * 4;      // edge weights (H<=4)
    float* el    = ws + o; o += (size_t)Nn * 4;
    float* er    = ws + o; o += (size_t)Nn * 4;
    float* mbuf  = ws + o; o += (size_t)Nn * 4;
    float* sbuf  = ws + o; o += (size_t)Nn * 4;
    float* W2pad = ws + o; o += (size_t)128 * 48;   // zero-padded W2 [128,48]

    // Pad W2 [128,40] -> [128,48] once per call (deterministic).
    gat_pad_w<<<(128 * 48 + 255) / 256, 256, 0, stream>>>(W2, W2pad, 128, 40, 48);

    auto edge_phase = [&](const float* feat, const float* al, const float* ar,
                          const float* bias, int H, int D, float* out, int relu) {
        const int HD  = H * D;
        const int NH  = Nn * H;
        const int NHD = Nn * HD;
        gat_node_attn<<<(NH + 255) / 256, 256, 0, stream>>>(feat, al, ar, el, er, Nn, H, D);
        gat_init<<<(NHD + 255) / 256, 256, 0, stream>>>(mbuf, sbuf, accb, NH, NHD);

        const int EH = E * H;
        gat_edge_max<<<(EH + 255) / 256, 256, 0, stream>>>(el, er, src, dst, mbuf, E, H);
        gat_edge_w<<<(EH + 255) / 256, 256, 0, stream>>>(el, er, mbuf, src, dst, wbuf, sbuf, E, H);

        const long long ED = (long long)E * D;
        gat_edge_agg<<<(int)((ED + 255) / 256), 256, 0, stream>>>(wbuf, feat, src, dst, accb, E, H, D);

        gat_node_finish<<<(NHD + 255) / 256, 256, 0, stream>>>(accb, sbuf, bias, out, Nn, H, D, relu);
    };

    const int mtiles = Nn / 16;                     // 3125
    const int gblk   = (mtiles * 32 + 127) / 128;   // 4 waves / block

    // Layer 0: [N,256] @ [256,128] -> [N,4,32]; ReLU
    gat_gemm_wmma<8><<<gblk, 128, 0, stream>>>(x, W0, featA, Nn, 256, 128);
    edge_phase(featA, al0, ar0, b0, 4, 32, featB, 1);

    // Layer 1: [N,128] @ [128,128] -> [N,4,32]; ReLU
    gat_gemm_wmma<8><<<gblk, 128, 0, stream>>>(featB, W1, featA, Nn, 128, 128);
    edge_phase(featA, al1, ar1, b1, 4, 32, featB, 1);

    // Layer 2: [N,128] @ [128,40(pad 48)] -> [N,1,40]; mean over 1 head == id
    gat_gemm_wmma<3><<<gblk, 128, 0, stream>>>(featB, W2pad, featA, Nn, 128, 40);
    edge_phase(featA, al2, ar2, b2, 1, 40, (float*)d_out, 0);
}